// FFCompatibility_48318382080005
// MI455X (gfx1250) — compile-verified
//
#include <hip/hip_runtime.h>
#include <hip/hip_bf16.h>
#include <stdint.h>

// ---------------------------------------------------------------------------
// FFCompatibility (DREAMPlace-style): truncated-Gaussian demand scatter into a
// 512x512x16x8 f32 map (128 MB, resident in MI455X's 192 MB L2), then a
// masked re-gather.
//
// Phase 0: zero demand map via TENSOR_STORE_FROM_LDS (TDM, iterate mode)
// Phase 1: per-flop 5x5 atomic scatter        (global_atomic_add_f32, L2)
// Phase 2: per-flop 5x5 masked gather + sum   (global_load_b32, L2)
// ---------------------------------------------------------------------------

#define NN      1200000   // total nodes
#define FF      1000000   // flops
constexpr int   NX  = 512;
constexpr int   NY  = 512;
constexpr int   NCK = 16;
constexpr int   NCE = 8;
constexpr int   CC  = NCK * NCE;            // 128 slots per (bx,by) cell
constexpr long long MAP_ELEMS = (long long)NX * NY * NCK * NCE; // 33,554,432
constexpr float XL = 0.0f, YL = 0.0f;
constexpr float SX = 1.0f, SY = 1.0f;
constexpr float INV_SX = 1.0f, INV_SY = 1.0f;
constexpr int   EXT = 2;                    // window = 2*EXT+1 = 5
constexpr int   W   = 2 * EXT + 1;
constexpr float SQRT2_INV = 0.70710678118654752440f;
constexpr float INV_SLICE_CAP = 1.0f / 16.0f;

// TDM zero-fill partition: 256 workgroups x (16 iterations x 32 KB) = 128 MB
constexpr int TDM_TILE_ELEMS   = 8192;                       // 32 KB LDS tile
constexpr int TDM_ITERS        = 16;                         // iterate_count+1
constexpr int TDM_REGION_ELEMS = TDM_TILE_ELEMS * TDM_ITERS; // 131072 / WG
constexpr int TDM_NBLOCKS      = (int)(MAP_ELEMS / TDM_REGION_ELEMS); // 256

// ---------------------------------------------------------------------------
// Fallback / small zero fill (grid-stride, 16B stores)
// ---------------------------------------------------------------------------
__global__ void ff_zero_f4(float4* __restrict__ p, int n4) {
    int i      = blockIdx.x * blockDim.x + threadIdx.x;
    int stride = gridDim.x * blockDim.x;
    float4 z = make_float4(0.0f, 0.0f, 0.0f, 0.0f);
    for (; i < n4; i += stride) p[i] = z;
}

// ---------------------------------------------------------------------------
// Phase 0: TDM zero fill. One wave zeroes a 32 KB LDS tile, then a single
// tensor_store_from_lds descriptor (iterate mode) streams it 16x into the
// workgroup's 512 KB slice of the map. Tracked by TENSORcnt.
//
// This toolchain's builtin is the 6-arg (clang-23 / therock-10.0) form:
//   (uint32x4 g0, int32x8 g1, int32x4 g2, int32x4 g3, int32x8 extra, i32 cpol)
// ---------------------------------------------------------------------------
#if __has_builtin(__builtin_amdgcn_tensor_store_from_lds)
#define FF_USE_TDM_ZERO 1
typedef unsigned int u32x4 __attribute__((ext_vector_type(4)));
typedef int          i32x8 __attribute__((ext_vector_type(8)));
typedef int          i32x4 __attribute__((ext_vector_type(4)));

__global__ void __launch_bounds__(32)
ff_zero_map_tdm(float* __restrict__ map) {
    __shared__ float tile[TDM_TILE_ELEMS];         // 32 KB of zeros

    // Zero the LDS tile: 2048 float4 across 32 lanes (64 b128 ds-stores/lane)
    float4* t4 = (float4*)tile;
    float4  z  = make_float4(0.0f, 0.0f, 0.0f, 0.0f);
    for (int i = threadIdx.x; i < TDM_TILE_ELEMS / 4; i += 32) t4[i] = z;
    asm volatile("s_wait_dscnt 0x0" ::: "memory");  // LDS writes visible to TDM

    unsigned lds_addr = (unsigned)(uintptr_t)&tile[0];  // low 32b = LDS offset
    unsigned long long gaddr =
        (unsigned long long)(uintptr_t)(map + (size_t)blockIdx.x * TDM_REGION_ELEMS);

    // ---- D# group 0 (CDNA5 ISA) ----
    u32x4 g0;
    g0.x = 0x1u;                                    // count=1, user descriptor
    g0.y = lds_addr;                                // lds_addr (bytes)
    g0.z = (unsigned)(gaddr & 0xFFFFFFFFull);       // global_addr[31:0]
    g0.w = (unsigned)((gaddr >> 32) & 0x01FFFFFFull)
         | 0x80000000u;                             // global_addr[56:32] | type=2

    // ---- D# group 1 (§8.4): data_size=4B, iterate_enable, 1-D tile ----
    const unsigned tdim0 = (unsigned)TDM_REGION_ELEMS;  // OOB guard per region
    const unsigned tdim1 = 1u;
    i32x8 g1;
    g1[0] = (2 << 16) | (1 << 19);                  // data_size=2(4B) | iterate
    g1[1] = (int)((tdim0 & 0xFFFFu) << 16);         // tensor_dim0[15:0]
    g1[2] = (int)(((tdim0 >> 16) & 0xFFFFu) | ((tdim1 & 0xFFFFu) << 16));
    g1[3] = (int)(((tdim1 >> 16) & 0xFFFFu) | ((unsigned)TDM_TILE_ELEMS << 16));
    g1[4] = 0;                                      // tile_dim1=0, tile_dim2=0
    g1[5] = (int)tdim0;                             // tensor_dim0_stride[31:0]
    g1[6] = 0;
    g1[7] = 0;

    // ---- D# group 2 (§8.5, iterate mode) ----
    i32x4 g2;
    g2[0] = 0;                                      // tensor_dim2 (unused)
    g2[1] = 0;                                      // lds_addr_increment = 0
    g2[2] = TDM_TILE_ELEMS;                         // global_addr_increment
    g2[3] = (TDM_ITERS - 1) << 16;                  // iterate_count = 15 -> 16x
    i32x4 g3 = {0, 0, 0, 0};

    // Extra descriptor group required by the 6-arg builtin variant (unused
    // here; zero-filled), cpol = 0 (default RT / device scope).
    i32x8 gx = {0, 0, 0, 0, 0, 0, 0, 0};

    __builtin_amdgcn_tensor_store_from_lds(g0, g1, g2, g3, gx, 0);
#if __has_builtin(__builtin_amdgcn_s_wait_tensorcnt)
    __builtin_amdgcn_s_wait_tensorcnt(0);
#else
    asm volatile("s_wait_tensorcnt 0x0" ::: "memory");
#endif
}
#else
#define FF_USE_TDM_ZERO 0
#endif

// ---------------------------------------------------------------------------
// Branch-free erf (Abramowitz-Stegun 7.1.26, |err| <= 1.5e-7). One v_exp_f32
// + FMA chain, no divergence (library erff has |x|<1 / tail branches).
// Arguments here are bounded to |x| <~ 2.2 (unit bins, +-2.5 sigma window).
// ---------------------------------------------------------------------------
__device__ __forceinline__ float ff_erff(float x) {
    float s    = copysignf(1.0f, x);
    float ax   = fabsf(x);
    float t    = __builtin_amdgcn_rcpf(fmaf(0.3275911f, ax, 1.0f));
    float poly = fmaf(t, 1.061405429f, -1.453152027f);
    poly = fmaf(t, poly, 1.421413741f);
    poly = fmaf(t, poly, -0.284496736f);
    poly = fmaf(t, poly, 0.254829592f);
    poly = poly * t;
    float e = __expf(-ax * ax);                    // v_exp_f32 path
    return s * fmaf(-poly, e, 1.0f);
}

// ---------------------------------------------------------------------------
// Per-axis truncated-Gaussian demand.
// 6 erf edge evaluations; normalization telescopes: sum(dem) = .5*(e[W]-e[0]).
// ---------------------------------------------------------------------------
__device__ __forceinline__ void ff_axis_demand(float c, float lo, float inv_std,
                                               float std_, int* b0_out,
                                               float dem[W]) {
    int b0 = (int)floorf((c - lo) * inv_std);
    *b0_out = b0;
    float e[W + 1];
#pragma unroll
    for (int j = 0; j <= W; ++j) {
        float left = lo + (float)(b0 + j - EXT) * std_;
        e[j] = ff_erff((left - c) * inv_std * SQRT2_INV);
    }
    float inv_sum = 1.0f / (0.5f * (e[W] - e[0]));   // telescoping sum
#pragma unroll
    for (int i = 0; i < W; ++i)
        dem[i] = 0.5f * (e[i + 1] - e[i]) * inv_sum;
}

// ---------------------------------------------------------------------------
// Phase 1: scatter 25 atomic contributions per flop into the demand map.
// Reference scatters at CLIPPED indices unconditionally (mask applies only to
// the gather), so no masking here.
// ---------------------------------------------------------------------------
__global__ void ff_scatter_kernel(const float* __restrict__ pos,
                                  const int*   __restrict__ flop_indices,
                                  const int*   __restrict__ flop_ctrl,   // [F,3]
                                  const float* __restrict__ nsx,
                                  const float* __restrict__ nsy,
                                  float*       __restrict__ dem_map) {
    int t = blockIdx.x * blockDim.x + threadIdx.x;
    if (t >= FF) return;

    int   fi = flop_indices[t];
    float cx = pos[fi]      + 0.5f * nsx[fi];
    float cy = pos[NN + fi] + 0.5f * nsy[fi];

    // & (pow2-1) == Python-style mod for any sign
    int cksr = flop_ctrl[3 * t + 1] & (NCK - 1);
    int ce   = flop_ctrl[3 * t + 2] & (NCE - 1);
    int slot = cksr * NCE + ce;

    int bx0, by0;
    float dx[W], dy[W];
    ff_axis_demand(cx, XL, INV_SX, SX, &bx0, dx);
    ff_axis_demand(cy, YL, INV_SY, SY, &by0, dy);

    int yoff[W];
#pragma unroll
    for (int j = 0; j < W; ++j) {
        int by = by0 - EXT + j;
        by = by < 0 ? 0 : (by > NY - 1 ? NY - 1 : by);
        yoff[j] = by * CC;
    }

#pragma unroll
    for (int i = 0; i < W; ++i) {
        int bx = bx0 - EXT + i;
        bx = bx < 0 ? 0 : (bx > NX - 1 ? NX - 1 : bx);
        float dxi = dx[i] * (SX * SY);
        float* row = dem_map + (size_t)bx * (NY * CC) + slot;
#pragma unroll
        for (int j = 0; j < W; ++j)
            atomicAdd(row + yoff[j], dxi * dy[j]);   // non-returning f32 atomic
    }
}

// ---------------------------------------------------------------------------
// Phase 2: masked gather of the same 25 cells; no erf needed here — only the
// bin indices and the in-range mask. Loads are unconditional (clipped addr),
// mask applied branchlessly; result accumulated into out with 1/SLICE_CAP.
// ---------------------------------------------------------------------------
__global__ void ff_gather_kernel(const float* __restrict__ pos,
                                 const int*   __restrict__ flop_indices,
                                 const int*   __restrict__ flop_ctrl,
                                 const float* __restrict__ nsx,
                                 const float* __restrict__ nsy,
                                 const float* __restrict__ dem_map,
                                 float*       __restrict__ out) {
    int t = blockIdx.x * blockDim.x + threadIdx.x;
    if (t >= FF) return;

    int   fi = flop_indices[t];
    float cx = pos[fi]      + 0.5f * nsx[fi];
    float cy = pos[NN + fi] + 0.5f * nsy[fi];

    int cksr = flop_ctrl[3 * t + 1] & (NCK - 1);
    int ce   = flop_ctrl[3 * t + 2] & (NCE - 1);
    int slot = cksr * NCE + ce;

    int bx0 = (int)floorf((cx - XL) * INV_SX);
    int by0 = (int)floorf((cy - YL) * INV_SY);

    int   yoff[W];
    float ymask[W];
#pragma unroll
    for (int j = 0; j < W; ++j) {
        int byu = by0 - EXT + j;
        ymask[j] = (byu >= 0 && byu < NY) ? 1.0f : 0.0f;
        int by = byu < 0 ? 0 : (byu > NY - 1 ? NY - 1 : byu);
        yoff[j] = by * CC;
    }

    float acc = 0.0f;
#pragma unroll
    for (int i = 0; i < W; ++i) {
        int bxu = bx0 - EXT + i;
        float xm = (bxu >= 0 && bxu < NX) ? 1.0f : 0.0f;
        int bx = bxu < 0 ? 0 : (bxu > NX - 1 ? NX - 1 : bxu);
        const float* row = dem_map + (size_t)bx * (NY * CC) + slot;
#pragma unroll
        for (int j = 0; j < W; ++j) {
            float v = row[yoff[j]];                  // always in-bounds (clipped)
            acc = fmaf(v, xm * ymask[j], acc);       // branchless mask
        }
    }

    atomicAdd(out + fi, acc * INV_SLICE_CAP);        // fi may repeat in general
}

// ---------------------------------------------------------------------------
// Launcher: zero(map via TDM) -> zero(out) -> scatter -> gather, stream-ordered.
// ---------------------------------------------------------------------------
extern "C" void kernel_launch(void* const* d_in, const int* in_sizes, int n_in,
                              void* d_out, int out_size, void* d_ws, size_t ws_size,
                              hipStream_t stream) {
    const float* pos          = (const float*)d_in[0];
    const int*   flop_indices = (const int*)  d_in[1];
    const int*   flop_ctrl    = (const int*)  d_in[2];
    const float* nsx          = (const float*)d_in[3];
    const float* nsy          = (const float*)d_in[4];
    float*       out          = (float*)d_out;
    float*       dem_map      = (float*)d_ws;

    const size_t map_bytes = (size_t)MAP_ELEMS * sizeof(float);
    if (ws_size < map_bytes) return;   // scratch too small: nothing safe to do

    const int threads = 256;

    // Phase 0: zero the 128 MB map + the output vector
#if FF_USE_TDM_ZERO
    ff_zero_map_tdm<<<TDM_NBLOCKS, 32, 0, stream>>>(dem_map);
#else
    {
        int n4 = (int)(MAP_ELEMS / 4);
        ff_zero_f4<<<4096, threads, 0, stream>>>((float4*)dem_map, n4);
    }
#endif
    {
        int on4     = out_size / 4;                   // N divisible by 4
        int oblocks = (on4 + threads - 1) / threads;
        ff_zero_f4<<<oblocks, threads, 0, stream>>>((float4*)out, on4);
    }

    // Phase 1: scatter
    {
        int blocks = (FF + threads - 1) / threads;
        ff_scatter_kernel<<<blocks, threads, 0, stream>>>(
            pos, flop_indices, flop_ctrl, nsx, nsy, dem_map);
    }

    // Phase 2: gather + reduce + output
    {
        int blocks = (FF + threads - 1) / threads;
        ff_gather_kernel<<<blocks, threads, 0, stream>>>(
            pos, flop_indices, flop_ctrl, nsx, nsy, dem_map, out);
    }
}